// HMCEN_NoHetGate_1855425872278
// MI455X (gfx1250) — compile-verified
//
#include <hip/hip_runtime.h>

#define N_NODES 100000
#define N_EDGES 1600000
#define IN_DIM 256
#define HID 128
#define FUSE 64
#define XW_W (2 * HID)  // concatenated [homo | het] = 256 columns

typedef float v2f __attribute__((ext_vector_type(2)));
typedef float v4f __attribute__((ext_vector_type(4)));
typedef float v8f __attribute__((ext_vector_type(8)));

__device__ __forceinline__ v8f wmma16x16x4(v2f a, v2f b, v8f c) {
  // D = A(16x4 f32) * B(4x16 f32) + C(16x16 f32), full fp32 precision
  return __builtin_amdgcn_wmma_f32_16x16x4_f32(false, a, false, b, (short)0, c,
                                               false, false);
}

// Wave-relative LDS byte offset of a generic pointer to __shared__ memory
// (ISA 10.2: generic->LDS mapping uses addr[31:0] as the DS address).
__device__ __forceinline__ unsigned lds_off(const void* p) {
  return (unsigned)(size_t)p;
}

// ---------------------------------------------------------------------------
// Stage 1: XW = x @ [W_homo | W_het]   (100000x256 @ 256x256, fp32 WMMA)
// Block = 32 rows x 256 cols, 8 waves. Wave w owns 32 columns
// (waves 0-3 -> W_homo, waves 4-7 -> W_het) and both 16-row A tiles.
// B panel (32 cols x 256 K) is staged wave-privately into LDS with
// double-buffered global_load_async_to_lds_b128 in 16-K-row chunks;
// per-wave s_wait_asynccnt pipelining, no barriers.
// ---------------------------------------------------------------------------
#define CHUNK_K 16
#define NCHUNK (IN_DIM / CHUNK_K)  // 16

__global__ __launch_bounds__(256) void k_xw_gemm(const float* __restrict__ x,
                                                 const float* __restrict__ Wh,
                                                 const float* __restrict__ Wt,
                                                 float* __restrict__ xw) {
  // [2 buffers][8 waves][16 K x 32 cols] floats = 32 KB
  __shared__ float smem[2 * 8 * CHUNK_K * 32];

  const int lane = threadIdx.x & 31;
  const int w = threadIdx.x >> 5;
  const int m = lane & 15;
  const int koff = (lane >> 4) << 1;  // 0 or 2
  const int rowBase = blockIdx.x << 5;
  const float* __restrict__ Wsel = (w < 4) ? Wh : Wt;
  const int cb0 = (w & 3) << 5;  // panel's first column within Wsel

  // A: two 16-row tiles of x
  const float* xrow0 = x + (size_t)(rowBase + m) * IN_DIM + koff;
  const float* xrow1 = xrow0 + (size_t)16 * IN_DIM;

  // async staging addresses: one b128 instr moves 4 K-rows x 32 cols (512B)
  const float* gbase =
      Wsel + (size_t)(lane >> 3) * HID + cb0 + ((lane & 7) << 2);
  const unsigned lbase = lds_off(smem) +
                         (unsigned)(w * (CHUNK_K * 32 * 4)) +
                         (unsigned)((lane >> 3) * 128 + ((lane & 7) << 4));

#define STAGE(buf_, kc_)                                                      \
  do {                                                                        \
    const float* g_ = gbase + (size_t)(kc_) * CHUNK_K * HID;                  \
    unsigned l_ = lbase + (unsigned)(buf_) * (8 * CHUNK_K * 32 * 4);          \
    asm volatile("global_load_async_to_lds_b128 %0, %1, off" ::"v"(l_),       \
                 "v"(g_)                                                      \
                 : "memory");                                                 \
    asm volatile("global_load_async_to_lds_b128 %0, %1, off" ::"v"(l_ + 512u),\
                 "v"(g_ + 4 * HID)                                            \
                 : "memory");                                                 \
    asm volatile("global_load_async_to_lds_b128 %0, %1, off" ::"v"(l_ + 1024u),\
                 "v"(g_ + 8 * HID)                                            \
                 : "memory");                                                 \
    asm volatile("global_load_async_to_lds_b128 %0, %1, off" ::"v"(l_ + 1536u),\
                 "v"(g_ + 12 * HID)                                           \
                 : "memory");                                                 \
  } while (0)

  v8f c00 = {0.f, 0.f, 0.f, 0.f, 0.f, 0.f, 0.f, 0.f};
  v8f c01 = c00, c10 = c00, c11 = c00;

  STAGE(0, 0);
  int buf = 0;
  for (int kc = 0; kc < NCHUNK; ++kc) {
    if (kc + 1 < NCHUNK) {
      STAGE(buf ^ 1, kc + 1);
      // 4 loads just issued; <=4 outstanding means chunk kc has landed
      asm volatile("s_wait_asynccnt 0x4" ::: "memory");
    } else {
      asm volatile("s_wait_asynccnt 0x0" ::: "memory");
    }
    const float* bpanel = smem + buf * (8 * CHUNK_K * 32) + w * (CHUNK_K * 32);
    const float* xp0 = xrow0 + kc * CHUNK_K;
    const float* xp1 = xrow1 + kc * CHUNK_K;
#pragma unroll
    for (int kk = 0; kk < CHUNK_K; kk += 4) {
      const int k = kk + koff;
      v2f a0 = *(const v2f*)(xp0 + kk);
      v2f a1 = *(const v2f*)(xp1 + kk);
      const float* bp = bpanel + k * 32 + m;
      v2f b0, b1;
      b0.x = bp[0];
      b0.y = bp[32];
      b1.x = bp[16];
      b1.y = bp[48];
      c00 = wmma16x16x4(a0, b0, c00);
      c01 = wmma16x16x4(a0, b1, c01);
      c10 = wmma16x16x4(a1, b0, c10);
      c11 = wmma16x16x4(a1, b1, c11);
    }
    buf ^= 1;
  }
#undef STAGE

  // C layout: VGPR i -> row base + i (+8 for upper half-wave), col = n0 + m
  const int gcb = ((w >= 4) ? HID : 0) + cb0 + m;
  const int rofs = (lane >> 4) << 3;
  float* op0 = xw + (size_t)(rowBase + rofs) * XW_W + gcb;
  float* op1 = xw + (size_t)(rowBase + 16 + rofs) * XW_W + gcb;
#pragma unroll
  for (int i = 0; i < 8; ++i) {
    op0[(size_t)i * XW_W] = c00[i];
    op0[(size_t)i * XW_W + 16] = c01[i];
    op1[(size_t)i * XW_W] = c10[i];
    op1[(size_t)i * XW_W + 16] = c11[i];
  }
}

// ---------------------------------------------------------------------------
// Stage 2: degree (dst-based, +1 self loop) and dinv = rsqrt(deg), in place
// ---------------------------------------------------------------------------
__global__ void k_deg_init(float* __restrict__ deg) {
  int n = blockIdx.x * blockDim.x + threadIdx.x;
  if (n < N_NODES) deg[n] = 1.0f;  // self loop
}

__global__ void k_deg_count(const long long* __restrict__ ei,
                            float* __restrict__ deg) {
  int e = blockIdx.x * blockDim.x + threadIdx.x;
  if (e < N_EDGES) atomicAdd(deg + (int)ei[N_EDGES + e], 1.0f);
}

__global__ void k_dinv(float* __restrict__ deg) {
  int n = blockIdx.x * blockDim.x + threadIdx.x;
  if (n < N_NODES) deg[n] = rsqrtf(deg[n]);  // deg >= 1 always
}

// ---------------------------------------------------------------------------
// Stage 3a: agg = xw * dinv[n]^2   (self-loop contribution, no atomics)
// ---------------------------------------------------------------------------
__global__ void k_agg_self(const float* __restrict__ xw,
                           const float* __restrict__ dinv,
                           float* __restrict__ agg) {
  int t = blockIdx.x * blockDim.x + threadIdx.x;  // over N * 64 float4
  if (t >= N_NODES * (XW_W / 4)) return;
  int n = t >> 6;
  float s = dinv[n];
  s = s * s;
  v4f v = ((const v4f*)xw)[t];
  v4f o;
  o.x = v.x * s;
  o.y = v.y * s;
  o.z = v.z * s;
  o.w = v.w * s;
  ((v4f*)agg)[t] = o;
}

// ---------------------------------------------------------------------------
// Stage 3b: edge scatter. One wave per edge: lane handles 8 of 256 dims.
// src/dst are wave-uniform -> readfirstlane keeps address math scalar.
// agg[dst] += xw[src] * dinv[src]*dinv[dst]   (f32 global atomics)
// ---------------------------------------------------------------------------
__global__ __launch_bounds__(256) void k_agg_edges(
    const long long* __restrict__ ei, const float* __restrict__ xw,
    const float* __restrict__ dinv, float* __restrict__ agg) {
  const int lane = threadIdx.x & 31;
  const int e = (blockIdx.x << 3) + (threadIdx.x >> 5);
  if (e >= N_EDGES) return;  // wave-uniform
  const int s = __builtin_amdgcn_readfirstlane((int)ei[e]);
  const int d = __builtin_amdgcn_readfirstlane((int)ei[N_EDGES + e]);
  const float nw = dinv[s] * dinv[d];
  const float* sp = xw + (size_t)s * XW_W + lane * 8;
  float* dp = agg + (size_t)d * XW_W + lane * 8;
  v4f v0 = *(const v4f*)(sp);
  v4f v1 = *(const v4f*)(sp + 4);
  atomicAdd(dp + 0, v0.x * nw);
  atomicAdd(dp + 1, v0.y * nw);
  atomicAdd(dp + 2, v0.z * nw);
  atomicAdd(dp + 3, v0.w * nw);
  atomicAdd(dp + 4, v1.x * nw);
  atomicAdd(dp + 5, v1.y * nw);
  atomicAdd(dp + 6, v1.z * nw);
  atomicAdd(dp + 7, v1.w * nw);
}

// ---------------------------------------------------------------------------
// Stage 4: fused readout. One wave per 16-node tile.
// h_fused = 0.5*(relu(aggH+bH)+relu(aggT+bT)); h = relu(h_fused@W_fuse+b_fuse)
// via 4 WMMA C-tiles (K=128); logits = h@W_cls+b_cls via shfl-xor reduction.
// ---------------------------------------------------------------------------
__global__ __launch_bounds__(256) void k_readout(
    const float* __restrict__ agg, const float* __restrict__ b_homo,
    const float* __restrict__ b_het, const float* __restrict__ W_fuse,
    const float* __restrict__ b_fuse, const float* __restrict__ W_cls,
    const float* __restrict__ b_cls, float* __restrict__ out) {
  const int gw = (int)((blockIdx.x * (unsigned)blockDim.x + threadIdx.x) >> 5);
  if (gw >= N_NODES / 16) return;  // wave-uniform exit before any WMMA
  const int lane = threadIdx.x & 31;
  const int m = lane & 15;
  const int koff = (lane >> 4) << 1;
  const float* arow = agg + (size_t)(gw * 16 + m) * XW_W + koff;

  v8f acc[4];
#pragma unroll
  for (int t = 0; t < 4; ++t)
    acc[t] = (v8f){0.f, 0.f, 0.f, 0.f, 0.f, 0.f, 0.f, 0.f};

#pragma unroll 2
  for (int k0 = 0; k0 < HID; k0 += 4) {
    const int k = k0 + koff;
    v2f ah = *(const v2f*)(arow + k0);        // homo half
    v2f at = *(const v2f*)(arow + k0 + HID);  // het half
    v2f bh = *(const v2f*)(b_homo + k);
    v2f bt = *(const v2f*)(b_het + k);
    v2f a;
    a.x = 0.5f * (fmaxf(ah.x + bh.x, 0.f) + fmaxf(at.x + bt.x, 0.f));
    a.y = 0.5f * (fmaxf(ah.y + bh.y, 0.f) + fmaxf(at.y + bt.y, 0.f));
    const float* wf = W_fuse + (size_t)k * FUSE + m;
#pragma unroll
    for (int t = 0; t < 4; ++t) {
      v2f b;
      b.x = wf[t * 16];
      b.y = wf[FUSE + t * 16];
      acc[t] = wmma16x16x4(a, b, acc[t]);
    }
  }

  // per-lane partial logits: this lane's column c = 16t+m, rows i (+8 upper)
  float p[16];
#pragma unroll
  for (int j = 0; j < 16; ++j) p[j] = 0.f;
#pragma unroll
  for (int t = 0; t < 4; ++t) {
    const int c = t * 16 + m;
    const float bf = b_fuse[c];
    const float wc0 = W_cls[c * 2 + 0];
    const float wc1 = W_cls[c * 2 + 1];
#pragma unroll
    for (int i = 0; i < 8; ++i) {
      float h = fmaxf(acc[t][i] + bf, 0.f);
      p[2 * i + 0] += h * wc0;
      p[2 * i + 1] += h * wc1;
    }
  }
  // reduce the 16 FUSE columns held across each 16-lane half (xor <= 8 stays
  // inside a half; lanes 0-15 hold rows 0-7, lanes 16-31 hold rows 8-15)
#pragma unroll
  for (int off = 1; off < 16; off <<= 1) {
#pragma unroll
    for (int j = 0; j < 16; ++j) p[j] += __shfl_xor(p[j], off, 32);
  }
  if (m == 0) {
    const float bc0 = b_cls[0];
    const float bc1 = b_cls[1];
    const int r0 = gw * 16 + ((lane >> 4) << 3);
#pragma unroll
    for (int i = 0; i < 8; ++i) {
      out[(size_t)(r0 + i) * 2 + 0] = p[2 * i + 0] + bc0;
      out[(size_t)(r0 + i) * 2 + 1] = p[2 * i + 1] + bc1;
    }
  }
}

// ---------------------------------------------------------------------------
extern "C" void kernel_launch(void* const* d_in, const int* in_sizes, int n_in,
                              void* d_out, int out_size, void* d_ws,
                              size_t ws_size, hipStream_t stream) {
  (void)in_sizes;
  (void)n_in;
  (void)out_size;
  (void)ws_size;
  const float* x = (const float*)d_in[0];
  const long long* ei = (const long long*)d_in[1];  // int64 [2, E]
  // d_in[2] = h_node: unused by the reference (no het gate)
  const float* W_homo = (const float*)d_in[3];
  const float* b_homo = (const float*)d_in[4];
  const float* W_het = (const float*)d_in[5];
  const float* b_het = (const float*)d_in[6];
  const float* W_fuse = (const float*)d_in[7];
  const float* b_fuse = (const float*)d_in[8];
  const float* W_cls = (const float*)d_in[9];
  const float* b_cls = (const float*)d_in[10];
  float* out = (float*)d_out;

  float* ws = (float*)d_ws;
  float* xw = ws;                              // N * 256
  float* agg = ws + (size_t)N_NODES * XW_W;    // N * 256
  float* dinv = agg + (size_t)N_NODES * XW_W;  // N (deg then dinv, in place)

  k_xw_gemm<<<N_NODES / 32, 256, 0, stream>>>(x, W_homo, W_het, xw);
  k_deg_init<<<(N_NODES + 255) / 256, 256, 0, stream>>>(dinv);
  k_deg_count<<<(N_EDGES + 255) / 256, 256, 0, stream>>>(ei, dinv);
  k_dinv<<<(N_NODES + 255) / 256, 256, 0, stream>>>(dinv);
  k_agg_self<<<(N_NODES * (XW_W / 4) + 255) / 256, 256, 0, stream>>>(xw, dinv,
                                                                     agg);
  k_agg_edges<<<N_EDGES / 8, 256, 0, stream>>>(ei, xw, dinv, agg);
  k_readout<<<(N_NODES / 16 + 7) / 8, 256, 0, stream>>>(
      agg, b_homo, b_het, W_fuse, b_fuse, W_cls, b_cls, out);
}